// Sparsity_42941083025412
// MI455X (gfx1250) — compile-verified
//
#include <hip/hip_runtime.h>

typedef float v4f __attribute__((ext_vector_type(4)));

namespace {
constexpr int kN   = 64;
constexpr int kC   = 512;
constexpr int kHW  = 56 * 56;          // 3136 floats per channel plane
constexpr int kG   = kC / 4;           // 128 channel quads
constexpr int kHW4 = kHW / 4;          // 784 float4 positions per plane
constexpr int kTPB = 256;              // 8 wave32 per block
constexpr int kPos = kN * kG * kHW4;   // 6,422,528 vec4 group-positions
constexpr int kTiles = kPos / kTPB;    // 25,088 (exact)
constexpr int kBlocks = 1568;
constexpr int kTilesPerBlock = kTiles / kBlocks;  // 16 (exact)
}  // namespace

__global__ __launch_bounds__(kTPB) void sparsity24_kernel(
    const float* __restrict__ x, float* __restrict__ y) {
  // 2 buffers x 4 channels x 256 lanes x float4 = 32 KB LDS (of 320 KB/WGP).
  __shared__ alignas(16) float lbuf[2][4][kTPB * 4];
  const int tid = threadIdx.x;

  // Per-lane LDS byte addresses for the async copies. On AMDGCN the low 32
  // bits of a generic pointer to LDS are exactly the DS byte address.
  unsigned lds_slot[2][4];
#pragma unroll
  for (int b = 0; b < 2; ++b)
#pragma unroll
    for (int c = 0; c < 4; ++c)
      lds_slot[b][c] = (unsigned)(unsigned long long)&lbuf[b][c][tid * 4];

  const int t0 = blockIdx.x * kTilesPerBlock;

  // Element offset of channel 0 of this lane's quad/position for a tile.
  auto tile_base = [&](int tile) -> unsigned {
    unsigned pos  = (unsigned)tile * kTPB + (unsigned)tid;
    unsigned hw4  = pos % kHW4;
    unsigned rest = pos / kHW4;
    unsigned g    = rest % kG;
    unsigned n    = rest / kG;
    return (n * kC + g * 4u) * kHW + hw4 * 4u;
  };

  // Issue 4 async b128 copies (one per channel plane) global -> LDS.
  auto issue = [&](int tile, int buf) {
    const unsigned base = tile_base(tile);
#pragma unroll
    for (int c = 0; c < 4; ++c) {
      unsigned long long ga = (unsigned long long)(x + base + c * kHW);
      asm volatile("global_load_async_to_lds_b128 %0, %1, off"
                   :
                   : "v"(lds_slot[buf][c]), "v"(ga)
                   : "memory");
    }
  };

  issue(t0, 0);  // prologue: prefetch first tile

  for (int i = 0; i < kTilesPerBlock; ++i) {
    if (i + 1 < kTilesPerBlock) {
      issue(t0 + i + 1, (i + 1) & 1);  // prefetch next tile into other buffer
      // 8 async ops outstanding; loads complete in order, so <=4 means the
      // current tile's 4 copies have landed in LDS.
      asm volatile("s_wait_asynccnt 4" : : : "memory");
    } else {
      asm volatile("s_wait_asynccnt 0" : : : "memory");
    }

    const int buf = i & 1;
    float v[4][4];  // [channel][w]
#pragma unroll
    for (int c = 0; c < 4; ++c)
      *(v4f*)&v[c][0] = *(const v4f*)&lbuf[buf][c][tid * 4];

    float o[4][4];
#pragma unroll
    for (int w = 0; w < 4; ++w) {
      const float a0 = __builtin_fabsf(v[0][w]);
      const float a1 = __builtin_fabsf(v[1][w]);
      const float a2 = __builtin_fabsf(v[2][w]);
      const float a3 = __builtin_fabsf(v[3][w]);
      // rank_i = #{j : a_j < a_i  or (a_j == a_i and j < i)}; drop ranks 0,1.
      const int r0 = (int)(a1 < a0) + (int)(a2 < a0) + (int)(a3 < a0);
      const int r1 = (int)(a0 <= a1) + (int)(a2 < a1) + (int)(a3 < a1);
      const int r2 = (int)(a0 <= a2) + (int)(a1 <= a2) + (int)(a3 < a2);
      const int r3 = (int)(a0 <= a3) + (int)(a1 <= a3) + (int)(a2 <= a3);
      o[0][w] = (r0 >= 2) ? v[0][w] : 0.0f;
      o[1][w] = (r1 >= 2) ? v[1][w] : 0.0f;
      o[2][w] = (r2 >= 2) ? v[2][w] : 0.0f;
      o[3][w] = (r3 >= 2) ? v[3][w] : 0.0f;
    }

    const unsigned base = tile_base(t0 + i);
#pragma unroll
    for (int c = 0; c < 4; ++c)
      __builtin_nontemporal_store(*(const v4f*)&o[c][0],
                                  (v4f*)(y + base + c * kHW));
  }
}

extern "C" void kernel_launch(void* const* d_in, const int* in_sizes, int n_in,
                              void* d_out, int out_size, void* d_ws,
                              size_t ws_size, hipStream_t stream) {
  (void)in_sizes; (void)n_in; (void)out_size; (void)d_ws; (void)ws_size;
  const float* x = (const float*)d_in[0];
  float* y = (float*)d_out;
  sparsity24_kernel<<<kBlocks, kTPB, 0, stream>>>(x, y);
}